// LM_MLE_forSpline_1288490189408
// MI455X (gfx1250) — compile-verified
//
#include <hip/hip_runtime.h>
#include <hip/hip_bf16.h>

typedef __attribute__((ext_vector_type(2))) float v2f;
typedef __attribute__((ext_vector_type(8))) float v8f;

#define BATCH 16384
#define NPTS  256
#define KP    11
#define WAVES 8

__global__ __launch_bounds__(32 * WAVES)
void lm_mle_spline_kernel(const float* __restrict__ cur,    // [B,K]
                          const float* __restrict__ mu,     // [B,N]
                          const float* __restrict__ jac,    // [B,N,K]
                          const float* __restrict__ smp,    // [B,N]
                          const float* __restrict__ lam,    // [B]
                          const float* __restrict__ prmm,   // [K,2]
                          float* __restrict__ out_new,      // [B,K]
                          float* __restrict__ out_scale) {  // [B,K]
    __shared__ float tile[WAVES][16 * 16];

    const int lane = threadIdx.x & 31;
    const int wave = threadIdx.x >> 5;
    const int b    = blockIdx.x * WAVES + wave;
    const int hi   = lane >> 4;   // half-wave: WMMA-K {0,1} vs {2,3}
    const int k    = lane & 15;   // matrix row (A) / column (B,D)

    // branch-free lane masks (computed once)
    const bool  inK  = (k < KP);
    const int   kc   = inK ? k : 0;           // clamped, always in-bounds
    const float bsel = (k == KP) ? 1.f : 0.f; // beta column selector

    const float* jacb = jac + (size_t)b * (NPTS * KP);
    const float* mub  = mu  + (size_t)b * NPTS;
    const float* smpb = smp + (size_t)b * NPTS;

    // D = A(16xN) * B(Nx16): cols 0..10 = alpha, col 11 = beta, rows 11..15 = 0
    v8f acc = {0.f, 0.f, 0.f, 0.f, 0.f, 0.f, 0.f, 0.f};

#pragma unroll 4
    for (int n = 0; n < NPTS; n += 4) {
        const int n0 = n + 2 * hi;
        // A operand: A[k][n0], A[k][n0+1]; unconditional loads, masked values
        float a0 = jacb[n0 * KP + kc];
        float a1 = jacb[(n0 + 1) * KP + kc];
        a0 = inK ? a0 : 0.f;
        a1 = inK ? a1 : 0.f;
        // per-n Poisson weights (broadcast float2 loads, 8B aligned: n0 even)
        float2 m2 = *(const float2*)(mub + n0);
        float2 s2 = *(const float2*)(smpb + n0);
        float im0 = __builtin_amdgcn_rcpf(fmaxf(m2.x, 1e-9f));
        float im1 = __builtin_amdgcn_rcpf(fmaxf(m2.y, 1e-9f));
        float si0 = s2.x * im0;
        float si1 = s2.y * im1;
        float af0 = si0 * im0;
        float af1 = si1 * im1;
        float bf0 = si0 - 1.0f;
        float bf1 = si1 - 1.0f;
        // B operand: col<11 = af*jac, col 11 = bf (a0==0 there), col 12..15 = 0
        float b0 = a0 * af0 + bsel * bf0;
        float b1 = a1 * af1 + bsel * bf1;
        v2f A  = {a0, a1};
        v2f Bv = {b0, b1};
        acc = __builtin_amdgcn_wmma_f32_16x16x4_f32(false, A, false, Bv,
                                                    (short)0, acc, false, false);
    }

    // scale[k] = sum_r alpha[r][k]^2  (rows 11..15 are exactly zero)
    float s = 0.f;
#pragma unroll
    for (int v = 0; v < 8; ++v) s += acc[v] * acc[v];
    s += __shfl_xor(s, 16, 32);
    const float scale_k = s;   // valid for k = lane&15 < 11 (both halves)

    // stage D into LDS: D row = vgpr + 8*hi, col = lane&15
#pragma unroll
    for (int v = 0; v < 8; ++v)
        tile[wave][(v + 8 * hi) * 16 + k] = acc[v];
    __syncthreads();

    // row-per-lane reload: lane r owns augmented row [alpha[r][0..10] | beta[r]]
    const int rr = lane & 15;
    const float4* rowp = (const float4*)&tile[wave][rr * 16];
    float4 c0 = rowp[0], c1 = rowp[1], c2 = rowp[2];
    float a_[12] = {c0.x, c0.y, c0.z, c0.w,
                    c1.x, c1.y, c1.z, c1.w,
                    c2.x, c2.y, c2.z, c2.w};

    // Marquardt damping on the diagonal: a[r][r] += lambda * scale[r]
    const float damp = lam[b] * scale_k;
#pragma unroll
    for (int j = 0; j < KP; ++j)
        a_[j] += (lane == j) ? damp : 0.f;

    // fully-unrolled Gauss-Jordan, pivot rows broadcast via shuffles
#pragma unroll
    for (int p = 0; p < KP; ++p) {
        float pr[12];
#pragma unroll
        for (int j = 0; j < 12; ++j) {
            if (j < p) continue;            // already eliminated, identity cols
            pr[j] = __shfl(a_[j], p, 32);
        }
        float pinv = 1.0f / pr[p];
        float f    = a_[p] * pinv;
        bool  isP  = (lane == p);
#pragma unroll
        for (int j = 0; j < 12; ++j) {
            if (j < p) continue;
            float upd = a_[j] - f * pr[j];
            float nrm = pr[j] * pinv;
            a_[j] = isP ? nrm : upd;
        }
    }

    float step = a_[11];                    // solution component r = lane
    step = (step != step) ? 0.1f : step;    // NaN -> 0.1 per reference

    if (lane < KP) {
        float nv = cur[(size_t)b * KP + lane] + step;
        nv = fmaxf(nv, prmm[2 * lane]);
        nv = fminf(nv, prmm[2 * lane + 1]);
        out_new[(size_t)b * KP + lane]   = nv;
        out_scale[(size_t)b * KP + lane] = scale_k;
    }
}

extern "C" void kernel_launch(void* const* d_in, const int* in_sizes, int n_in,
                              void* d_out, int out_size, void* d_ws, size_t ws_size,
                              hipStream_t stream) {
    const float* cur  = (const float*)d_in[0];
    const float* mu   = (const float*)d_in[1];
    const float* jac  = (const float*)d_in[2];
    const float* smp  = (const float*)d_in[3];
    const float* lam  = (const float*)d_in[4];
    const float* prmm = (const float*)d_in[5];

    float* out_new   = (float*)d_out;
    float* out_scale = out_new + (size_t)BATCH * KP;

    dim3 grid(BATCH / WAVES);
    dim3 block(32 * WAVES);
    hipLaunchKernelGGL(lm_mle_spline_kernel, grid, block, 0, stream,
                       cur, mu, jac, smp, lam, prmm, out_new, out_scale);
}